// LmpGradOut_10660108828926
// MI455X (gfx1250) — compile-verified
//
#include <hip/hip_runtime.h>
#include <hip/hip_bf16.h>
#include <math.h>

// ---------------------------------------------------------------------------
// LmpGradOut: energies / forces / virials for a factored GNN readout.
//   N=200000 nodes, E=3200000 edges, D=128, H=64.
// Restructured:
//   vsum[n]   = sum_{e: recv=n} vec_e                     (3 atomics/edge)
//   x         = x_scalar + coord@Wc + vsum@Wv             (node)
//   z = x@W1+b1 ; h=silu(z) ; energy = h@W2+b2            (WMMA f32 16x16x4)
//   dh = W2 * silu'(z) ; g = dh@W1^T                      (WMMA f32 16x16x4)
//   forces = -g@Wc^T ; gv = g@Wv^T
//   virials[n] = 0.5*outer(gv[n],vsum[n])                 (receiver half, exact)
//              + scatter_{e: send=n} 0.5*outer(gv[r_e],vec_e)  (9 atomics/edge)
// ---------------------------------------------------------------------------

typedef __attribute__((ext_vector_type(2))) float v2f;
typedef __attribute__((ext_vector_type(8))) float v8f;

#define DIMS 128
#define HID  64
#define NPB  64   // nodes per block (4 waves x 16 nodes)

__global__ __launch_bounds__(256) void zero_f32_kernel(float* __restrict__ p, int n) {
    int i = blockIdx.x * blockDim.x + threadIdx.x;
    if (i < n) p[i] = 0.0f;
}

// vsum[r] += vec[e]  (3 float atomics per edge)
__global__ __launch_bounds__(256) void edge_vsum_kernel(const float* __restrict__ vec,
                                                        const int* __restrict__ recv,
                                                        float* __restrict__ vsum, int E) {
    int e = blockIdx.x * blockDim.x + threadIdx.x;
    if (e >= E) return;
    int r = recv[e];
    float v0 = vec[3 * (size_t)e + 0];
    float v1 = vec[3 * (size_t)e + 1];
    float v2 = vec[3 * (size_t)e + 2];
    atomicAdd(&vsum[3 * (size_t)r + 0], v0);
    atomicAdd(&vsum[3 * (size_t)r + 1], v1);
    atomicAdd(&vsum[3 * (size_t)r + 2], v2);
}

// Node MLP fwd + analytic bwd. One block = 64 nodes = 4 waves x (16-node WMMA tile).
__global__ __launch_bounds__(128) void node_mlp_kernel(
    const float* __restrict__ xs, const float* __restrict__ coord,
    const float* __restrict__ Wc, const float* __restrict__ Wv,
    const float* __restrict__ W1, const float* __restrict__ b1,
    const float* __restrict__ W2, const float* __restrict__ b2,
    const float* __restrict__ vsum,
    float* __restrict__ energies, float* __restrict__ forces,
    float* __restrict__ virials, float* __restrict__ gv_out, int n_nodes) {
    // +1 padding breaks the stride-128/64 LDS bank aliasing in the per-node passes.
    __shared__ float sX[NPB][DIMS + 1];  // x tile, later reused for g tile
    __shared__ float sZ[NPB][HID + 1];   // z tile, later reused for dh tile
    __shared__ float sC[NPB][3];         // coord tile
    __shared__ float sS[NPB][3];         // vsum tile

    const int tid  = threadIdx.x;
    const int base = blockIdx.x * NPB;

    // --- stage coord / vsum ------------------------------------------------
    for (int i = tid; i < NPB * 3; i += 128) {
        int m = i / 3, j = i % 3, gn = base + m;
        float c = 0.f, s = 0.f;
        if (gn < n_nodes) { c = coord[3 * (size_t)gn + j]; s = vsum[3 * (size_t)gn + j]; }
        sC[m][j] = c; sS[m][j] = s;
    }
    __syncthreads();

    // --- phase A: x = x_scalar + coord@Wc + vsum@Wv -> sX -------------------
    {
        int d = tid;  // 0..127, one feature column per thread
        float wc0 = Wc[d], wc1 = Wc[DIMS + d], wc2 = Wc[2 * DIMS + d];
        float wv0 = Wv[d], wv1 = Wv[DIMS + d], wv2 = Wv[2 * DIMS + d];
        for (int m = 0; m < NPB; ++m) {
            int gn = base + m;
            float v = 0.f;
            if (gn < n_nodes)
                v = xs[(size_t)gn * DIMS + d]
                  + sC[m][0] * wc0 + sC[m][1] * wc1 + sC[m][2] * wc2
                  + sS[m][0] * wv0 + sS[m][1] * wv1 + sS[m][2] * wv2;
            sX[m][d] = v;
        }
    }
    __syncthreads();

    const int wave = tid >> 5;       // 0..3
    const int l    = tid & 31;
    const int hlf  = l >> 4;         // lane half: K offset of 2 (ISA 32-bit A/B layout)
    const int ln   = l & 15;         // A row / B,C column
    const int m0   = wave * 16;      // wave's 16-node tile

    // --- phase B: forward z = x @ W1  (WMMA f32 16x16x4, K-loop over 128) ---
    for (int t = 0; t < HID / 16; ++t) {
        v8f acc = {0.f, 0.f, 0.f, 0.f, 0.f, 0.f, 0.f, 0.f};
        for (int kk = 0; kk < DIMS / 4; ++kk) {
            int kb = kk * 4 + hlf * 2;
            v2f a, b;
            a.x = sX[m0 + ln][kb];
            a.y = sX[m0 + ln][kb + 1];
            b.x = W1[(size_t)(kb)     * HID + t * 16 + ln];
            b.y = W1[(size_t)(kb + 1) * HID + t * 16 + ln];
            acc = __builtin_amdgcn_wmma_f32_16x16x4_f32(false, a, false, b,
                                                        (short)0, acc, false, false);
        }
        // C layout: vgpr j, lane-half h -> row j+8h, col = lane%16
        for (int j = 0; j < 8; ++j)
            sZ[m0 + j + 8 * hlf][t * 16 + ln] = acc[j];
    }
    __syncthreads();

    // --- phase C: per-node silu + energy + dh (overwrite sZ) ----------------
    if (tid < NPB) {
        int gn = base + tid;
        if (gn < n_nodes) {
            float e = b2[0];
            for (int k = 0; k < HID; ++k) {
                float z = sZ[tid][k] + b1[k];
                float s = 1.f / (1.f + expf(-z));
                e += (z * s) * W2[k];
                // d silu/dz = s*(1 + z*(1-s))
                sZ[tid][k] = W2[k] * s * (1.f + z * (1.f - s));
            }
            energies[gn] = e;
        }
    }
    __syncthreads();

    // --- phase D: backward g = dh @ W1^T  (WMMA, K-loop over 64) ------------
    for (int t = 0; t < DIMS / 16; ++t) {
        v8f acc = {0.f, 0.f, 0.f, 0.f, 0.f, 0.f, 0.f, 0.f};
        for (int kk = 0; kk < HID / 4; ++kk) {
            int kb = kk * 4 + hlf * 2;
            v2f a, b;
            a.x = sZ[m0 + ln][kb];
            a.y = sZ[m0 + ln][kb + 1];
            b.x = W1[(size_t)(t * 16 + ln) * HID + kb];      // W1^T[k][d] = W1[d][k]
            b.y = W1[(size_t)(t * 16 + ln) * HID + kb + 1];
            acc = __builtin_amdgcn_wmma_f32_16x16x4_f32(false, a, false, b,
                                                        (short)0, acc, false, false);
        }
        for (int j = 0; j < 8; ++j)
            sX[m0 + j + 8 * hlf][t * 16 + ln] = acc[j];      // reuse sX for g
    }
    __syncthreads();

    // --- phase E: forces, gv, receiver-half virials -------------------------
    if (tid < NPB) {
        int gn = base + tid;
        if (gn < n_nodes) {
            float f0 = 0.f, f1 = 0.f, f2 = 0.f, g0 = 0.f, g1 = 0.f, g2 = 0.f;
            for (int d = 0; d < DIMS; ++d) {
                float g = sX[tid][d];
                f0 += g * Wc[d];  f1 += g * Wc[DIMS + d];  f2 += g * Wc[2 * DIMS + d];
                g0 += g * Wv[d];  g1 += g * Wv[DIMS + d];  g2 += g * Wv[2 * DIMS + d];
            }
            forces[3 * (size_t)gn + 0] = -f0;
            forces[3 * (size_t)gn + 1] = -f1;
            forces[3 * (size_t)gn + 2] = -f2;
            gv_out[3 * (size_t)gn + 0] = g0;
            gv_out[3 * (size_t)gn + 1] = g1;
            gv_out[3 * (size_t)gn + 2] = g2;
            float s0 = sS[tid][0], s1 = sS[tid][1], s2 = sS[tid][2];
            float* vr = virials + 9 * (size_t)gn;
            vr[0] = 0.5f * g0 * s0; vr[1] = 0.5f * g0 * s1; vr[2] = 0.5f * g0 * s2;
            vr[3] = 0.5f * g1 * s0; vr[4] = 0.5f * g1 * s1; vr[5] = 0.5f * g1 * s2;
            vr[6] = 0.5f * g2 * s0; vr[7] = 0.5f * g2 * s1; vr[8] = 0.5f * g2 * s2;
        }
    }
}

// virials[send] += 0.5 * outer(gv[recv], vec_e)   (sender half; 9 atomics/edge)
__global__ __launch_bounds__(256) void edge_virial_kernel(const float* __restrict__ vec,
                                                          const int* __restrict__ ei,
                                                          const float* __restrict__ gv,
                                                          float* __restrict__ virials,
                                                          int E) {
    int e = blockIdx.x * blockDim.x + threadIdx.x;
    if (e >= E) return;
    int s = ei[e];
    int r = ei[(size_t)E + e];
    float g0 = gv[3 * (size_t)r + 0], g1 = gv[3 * (size_t)r + 1], g2 = gv[3 * (size_t)r + 2];
    float v0 = vec[3 * (size_t)e + 0], v1 = vec[3 * (size_t)e + 1], v2 = vec[3 * (size_t)e + 2];
    float* vr = virials + 9 * (size_t)s;
    atomicAdd(&vr[0], 0.5f * g0 * v0); atomicAdd(&vr[1], 0.5f * g0 * v1); atomicAdd(&vr[2], 0.5f * g0 * v2);
    atomicAdd(&vr[3], 0.5f * g1 * v0); atomicAdd(&vr[4], 0.5f * g1 * v1); atomicAdd(&vr[5], 0.5f * g1 * v2);
    atomicAdd(&vr[6], 0.5f * g2 * v0); atomicAdd(&vr[7], 0.5f * g2 * v1); atomicAdd(&vr[8], 0.5f * g2 * v2);
}

extern "C" void kernel_launch(void* const* d_in, const int* in_sizes, int n_in,
                              void* d_out, int out_size, void* d_ws, size_t ws_size,
                              hipStream_t stream) {
    const float* xs    = (const float*)d_in[0];
    const float* coord = (const float*)d_in[1];
    const float* vec   = (const float*)d_in[2];
    const int*   ei    = (const int*)d_in[3];   // [2, E] row-major
    const float* Wc    = (const float*)d_in[4];
    const float* Wv    = (const float*)d_in[5];
    const float* W1    = (const float*)d_in[6];
    const float* b1    = (const float*)d_in[7];
    const float* W2    = (const float*)d_in[8];
    const float* b2    = (const float*)d_in[9];

    const int N = in_sizes[0] / DIMS;
    const int E = in_sizes[2] / 3;

    float* out      = (float*)d_out;
    float* energies = out;
    float* forces   = out + (size_t)N;
    float* virials  = out + (size_t)N * 4;   // N + 3N

    float* vsum = (float*)d_ws;              // [N,3]
    float* gv   = vsum + (size_t)N * 3;      // [N,3]

    const int zn = N * 3;
    zero_f32_kernel<<<(zn + 255) / 256, 256, 0, stream>>>(vsum, zn);
    edge_vsum_kernel<<<(E + 255) / 256, 256, 0, stream>>>(vec, ei + (size_t)E, vsum, E);
    node_mlp_kernel<<<(N + NPB - 1) / NPB, 128, 0, stream>>>(
        xs, coord, Wc, Wv, W1, b1, W2, b2, vsum,
        energies, forces, virials, gv, N);
    edge_virial_kernel<<<(E + 255) / 256, 256, 0, stream>>>(vec, ei, gv, virials, E);
}